// GCRANE_58789512348195
// MI455X (gfx1250) — compile-verified
//
#include <hip/hip_runtime.h>

// SpMM (gather + segment-sum) on MI455X.
// Memory-bound (AI ~ 0.25 FLOP/B): no WMMA — optimize the data-movement path:
//  - b128 vector gathers, s_load edge metadata, native global_atomic_add_f32,
//  - global_prefetch_b8 for gather latency, NT stores for write-once region,
//  - keep the ~169MB hot set (x1 sources + atomic dest) resident in 192MB L2.

typedef float v4f __attribute__((ext_vector_type(4)));

#define DFEAT 128

// ---------------------------------------------------------------------------
// Scatter-SpMM: one wave32 per edge. lane l handles features [4l, 4l+4).
//   - edge index forced into SGPR (readfirstlane) so rows/cols/vals are s_load
//   - 16B global_load_b128 gather of the source row slice
//   - packed f32 multiply, then 4 native global_atomic_add_f32 in SADDR form:
//     SGPR base (out) + 32-bit lane offset + immediate offsets 0/4/8/12
//   - prefetch the next edge's source row (global_prefetch_b8) to hide the
//     random-gather latency across the grid-stride loop
// ---------------------------------------------------------------------------
__global__ __launch_bounds__(256) void spmm_scatter_kernel(
    const int* __restrict__ rows, const int* __restrict__ cols,
    const float* __restrict__ vals,
    const float* __restrict__ node, const float* __restrict__ attri,
    float* __restrict__ out, int nedges, int nnode) {
  const int lane = threadIdx.x & 31;
  const int wave = blockIdx.x * (256 / 32) + (threadIdx.x >> 5);
  const int stride = gridDim.x * (256 / 32);

  for (int ew = wave; ew < nedges; ew += stride) {
    const int e = __builtin_amdgcn_readfirstlane(ew);

    // Prefetch next iteration's gather source into the cache hierarchy.
    const int en = e + stride;
    if (en < nedges) {
      const int cn = cols[en];
      const float* pn = (cn < nnode) ? (node + (size_t)cn * DFEAT)
                                     : (attri + (size_t)(cn - nnode) * DFEAT);
      __builtin_prefetch(pn + lane * 4, 0, 3);
    }

    const int   r = rows[e];
    const int   c = cols[e];
    const float v = vals[e];

    const float* src = (c < nnode) ? (node + (size_t)c * DFEAT)
                                   : (attri + (size_t)(c - nnode) * DFEAT);
    const v4f x = *(const v4f*)(src + lane * 4);   // global_load_b128
    const v4f p = x * v;                           // packed f32 muls

    // 32-bit byte offset into out (max N*D*4 = 56.3MB < 2^31).
    const unsigned voff = ((unsigned)r * (DFEAT * 4u)) + (unsigned)(lane * 16);
    asm volatile(
        "global_atomic_add_f32 %0, %2, %1\n\t"
        "global_atomic_add_f32 %0, %3, %1 offset:4\n\t"
        "global_atomic_add_f32 %0, %4, %1 offset:8\n\t"
        "global_atomic_add_f32 %0, %5, %1 offset:12"
        :
        : "v"(voff), "s"(out), "v"(p.x), "v"(p.y), "v"(p.z), "v"(p.w)
        : "memory");
  }
}

// ---------------------------------------------------------------------------
// Init pass: out[0:N*D) = concat(node, attri)  (non-temporal: write-once,
//            never re-read on device -> don't burn L2 capacity on it)
//            out[N*D:3*N*D) = 0  (regular stores: these lines are about to be
//            hammered by L2 atomics, keep them resident/dirty)
// ---------------------------------------------------------------------------
__global__ __launch_bounds__(256) void init_out_kernel(
    const v4f* __restrict__ node, const v4f* __restrict__ attri,
    v4f* __restrict__ out, int nd4, int node4) {
  int i = blockIdx.x * blockDim.x + threadIdx.x;
  if (i >= 3 * nd4) return;
  if (i < nd4) {
    v4f val = (i < node4) ? node[i] : attri[i - node4];
    __builtin_nontemporal_store(val, &out[i]);
  } else {
    v4f z = {0.f, 0.f, 0.f, 0.f};
    out[i] = z;
  }
}

extern "C" void kernel_launch(void* const* d_in, const int* in_sizes, int n_in,
                              void* d_out, int out_size, void* d_ws, size_t ws_size,
                              hipStream_t stream) {
  const float* node   = (const float*)d_in[0];
  const float* attri  = (const float*)d_in[1];
  const int*   a_rows = (const int*)d_in[2];
  const int*   a_cols = (const int*)d_in[3];
  const float* a_vals = (const float*)d_in[4];
  const int*   b_rows = (const int*)d_in[5];
  const int*   b_cols = (const int*)d_in[6];
  const float* b_vals = (const float*)d_in[7];
  float* out = (float*)d_out;

  const int nnode  = in_sizes[0] / DFEAT;
  const int nattri = in_sizes[1] / DFEAT;
  const int n      = nnode + nattri;
  const int e1     = in_sizes[2];
  const int e2     = in_sizes[5];
  const int nd     = n * DFEAT;
  const int nd4    = nd / 4;
  const int node4  = (nnode * DFEAT) / 4;

  // Pass 1: x1 copy + zero accumulators (d_out is poisoned each time).
  const int init_blocks = (3 * nd4 + 255) / 256;
  init_out_kernel<<<init_blocks, 256, 0, stream>>>(
      (const v4f*)node, (const v4f*)attri, (v4f*)out, nd4, node4);

  // Passes 2/3: scatter-SpMM, ~8 edges per wave via grid-stride loop.
  auto spmm = [&](const int* r, const int* c, const float* v, int ne, float* o) {
    int waves  = (ne + 7) / 8;          // target ~8 iterations per wave
    int blocks = (waves + 7) / 8;       // 8 waves (256 threads) per block
    if (blocks < 1) blocks = 1;
    spmm_scatter_kernel<<<blocks, 256, 0, stream>>>(r, c, v, node, attri, o,
                                                    ne, nnode);
  };
  spmm(a_rows, a_cols, a_vals, e1, out + (size_t)nd);
  spmm(b_rows, b_cols, b_vals, e2, out + 2 * (size_t)nd);
}